// RMRM_6133213299512
// MI455X (gfx1250) — compile-verified
//
#include <hip/hip_runtime.h>
#include <math.h>

typedef __attribute__((ext_vector_type(2))) float v2f;
typedef __attribute__((ext_vector_type(8))) float v8f;

#define NEG_SLOPE 0.2f

// ---------------------------------------------------------------------------
// f32 GEMM via V_WMMA_F32_16X16X4_F32: C[M,N] = A[M,K] * B[K,N]  (row-major)
// One wave computes MT consecutive 16x16 m-tiles for a single n-tile, so each
// B fragment is reused MT times. Requires N%16==0, K%4==0. Ragged M handled by
// *uniform* clamping of the tile index (no lane divergence; EXEC stays all-1s
// around WMMA per ISA 7.12 restrictions). Layout per CDNA5 ISA 7.12.2:
//   A frag (v2f): lane L -> A[m0 + L%16][k0 + 2*(L/16) + r]
//   B frag (v2f): lane L -> B[k0 + 2*(L/16) + r][n0 + L%16]
//   C/D   (v8f): lane L, reg g -> C[m0 + g + 8*(L/16)][n0 + L%16]
// ---------------------------------------------------------------------------
#define GEMM_MT 4

__global__ void wmma_gemm_f32(const float* __restrict__ A,
                              const float* __restrict__ B,
                              float* __restrict__ C,
                              int M, int N, int K) {
  int gid  = blockIdx.x * blockDim.x + threadIdx.x;
  int wave = gid >> 5;
  int lane = threadIdx.x & 31;
  int nT   = N >> 4;
  int mT   = (M + 15) >> 4;
  int mG   = (mT + GEMM_MT - 1) / GEMM_MT;
  if (wave >= mG * nT) return;              // wave-uniform exit
  int grp = wave / nT;
  int nt  = wave - grp * nT;
  int mtBase = grp * GEMM_MT;
  int n0 = nt << 4;
  int half = lane >> 4;                     // 0: lanes 0-15, 1: lanes 16-31
  int idx  = lane & 15;

  // uniform per-tile row bases (clamped: out-of-range tiles compute garbage
  // into dead accumulators but never touch out-of-bounds memory)
  int rowBase[GEMM_MT];
#pragma unroll
  for (int t = 0; t < GEMM_MT; ++t) {
    int mt = mtBase + t;
    if (mt > mT - 1) mt = mT - 1;
    rowBase[t] = (mt << 4) + idx;           // row this lane reads/writes
  }

  v8f acc[GEMM_MT];
#pragma unroll
  for (int t = 0; t < GEMM_MT; ++t) acc[t] = (v8f){0,0,0,0,0,0,0,0};

  for (int k0 = 0; k0 < K; k0 += 4) {
    int ka = k0 + 2 * half;
    v2f b;
    b.x = B[ka * N + n0 + idx];
    b.y = B[(ka + 1) * N + n0 + idx];
#pragma unroll
    for (int t = 0; t < GEMM_MT; ++t) {
      v2f a = *(const v2f*)(A + (size_t)rowBase[t] * K + ka); // 8B-aligned b64
      acc[t] = __builtin_amdgcn_wmma_f32_16x16x4_f32(
          false, a, false, b, (short)0, acc[t], false, false);
    }
  }

#pragma unroll
  for (int t = 0; t < GEMM_MT; ++t) {
    if (mtBase + t >= mT) break;            // uniform
    int rb = ((mtBase + t) << 4) + 8 * half;
#pragma unroll
    for (int g = 0; g < 8; ++g) {
      int r = rb + g;
      if (r < M) C[(size_t)r * N + n0 + idx] = acc[t][g];
    }
  }
}

// ---------------------------------------------------------------------------
// Per-node attention coefficients: a_src[n,h] = sum_c xl[n,h,c]*att_src[h,c]
// ---------------------------------------------------------------------------
template <int H, int Cc>
__global__ void attn_coef(const float* __restrict__ xl,
                          const float* __restrict__ att_src,
                          const float* __restrict__ att_dst,
                          float* __restrict__ a_src,
                          float* __restrict__ a_dst,
                          int Nn) {
  int i = blockIdx.x * blockDim.x + threadIdx.x;
  if (i >= Nn * H) return;
  int n = i / H, h = i - n * H;
  const float* xp = xl + ((size_t)n * H + h) * Cc;
  const float* sp = att_src + h * Cc;
  const float* dp = att_dst + h * Cc;
  float s = 0.f, d = 0.f;
#pragma unroll
  for (int c = 0; c < Cc; ++c) {
    float v = xp[c];
    s += v * sp[c];
    d += v * dp[c];
  }
  a_src[i] = s;
  a_dst[i] = d;
}

// ---------------------------------------------------------------------------
// Utility kernels
// ---------------------------------------------------------------------------
__global__ void fill_f32(float* __restrict__ p, float v, int n) {
  int i = blockIdx.x * blockDim.x + threadIdx.x;
  if (i < n) p[i] = v;
}

template <int F>
__global__ void init_bias(float* __restrict__ out, const float* __restrict__ b,
                          int Nn) {
  int i = blockIdx.x * blockDim.x + threadIdx.x;
  if (i < Nn * F) out[i] = b[i & (F - 1)];
}

__global__ void elu_inplace(float* __restrict__ p, int n) {
  int i = blockIdx.x * blockDim.x + threadIdx.x;
  if (i < n) {
    float v = p[i];
    p[i] = v > 0.f ? v : (expf(v) - 1.f);
  }
}

// ---------------------------------------------------------------------------
// Edge helpers (self-loops appended: edge id >= E -> loop on node (e - E))
// edge_index is [2,E] row-major.
// ---------------------------------------------------------------------------
__device__ __forceinline__ void load_edge(const int* __restrict__ ei, int E,
                                          int e, int& s, int& d) {
  if (e < E) { s = ei[e]; d = ei[E + e]; }
  else       { s = e - E; d = e - E; }
}

__device__ __forceinline__ float lrelu(float v) {
  return v > 0.f ? v : NEG_SLOPE * v;
}

// bit-trick float atomic max (correct for mixed signs; init with -inf)
__device__ __forceinline__ void atomicMaxFloat(float* addr, float val) {
  if (val >= 0.f) atomicMax((int*)addr, __float_as_int(val));
  else            atomicMin((unsigned int*)addr, __float_as_uint(val));
}

// Pass A: segment max of leaky-relu'd logits over destination nodes
template <int H>
__global__ void edge_max(const int* __restrict__ ei, int E, int Etot,
                         const float* __restrict__ a_src,
                         const float* __restrict__ a_dst,
                         float* __restrict__ maxv) {
  int i = blockIdx.x * blockDim.x + threadIdx.x;
  if (i >= Etot * H) return;
  int e = i / H, h = i - e * H;
  int s, d; load_edge(ei, E, e, s, d);
  float v = lrelu(a_src[s * H + h] + a_dst[d * H + h]);
  atomicMaxFloat(&maxv[d * H + h], v);
}

// Pass B: denom[d,h] += exp(e - max)
template <int H>
__global__ void edge_denom(const int* __restrict__ ei, int E, int Etot,
                           const float* __restrict__ a_src,
                           const float* __restrict__ a_dst,
                           const float* __restrict__ maxv,
                           float* __restrict__ denom) {
  int i = blockIdx.x * blockDim.x + threadIdx.x;
  if (i >= Etot * H) return;
  int e = i / H, h = i - e * H;
  int s, d; load_edge(ei, E, e, s, d);
  float v = lrelu(a_src[s * H + h] + a_dst[d * H + h]);
  float ex = expf(v - maxv[d * H + h]);
  atomicAdd(&denom[d * H + h], ex);
}

// Pass C: out[d, h, c] += alpha(e,h) * xl[s, h, c]   (thread per edge-element)
template <int H, int Cc>
__global__ void edge_aggregate(const int* __restrict__ ei, int E, int Etot,
                               const float* __restrict__ a_src,
                               const float* __restrict__ a_dst,
                               const float* __restrict__ maxv,
                               const float* __restrict__ denom,
                               const float* __restrict__ xl,
                               float* __restrict__ out) {
  constexpr int HC = H * Cc;
  int i = blockIdx.x * blockDim.x + threadIdx.x;
  if (i >= Etot * HC) return;
  int e = i / HC, r = i - e * HC;
  int h = r / Cc;
  int s, d; load_edge(ei, E, e, s, d);
  float v = lrelu(a_src[s * H + h] + a_dst[d * H + h]);
  float ex = expf(v - maxv[d * H + h]);
  float alpha = ex / denom[d * H + h];
  atomicAdd(&out[(size_t)d * HC + r], xl[(size_t)s * HC + r] * alpha);
}

// ---------------------------------------------------------------------------
extern "C" void kernel_launch(void* const* d_in, const int* in_sizes, int n_in,
                              void* d_out, int out_size, void* d_ws, size_t ws_size,
                              hipStream_t stream) {
  const float* x    = (const float*)d_in[0];
  const int*   ei   = (const int*)  d_in[1];   // edge_index [2,E]
  const float* W1   = (const float*)d_in[2];
  const float* as1  = (const float*)d_in[3];
  const float* ad1  = (const float*)d_in[4];
  const float* b1   = (const float*)d_in[5];
  const float* W2   = (const float*)d_in[6];
  const float* as2  = (const float*)d_in[7];
  const float* ad2  = (const float*)d_in[8];
  const float* b2   = (const float*)d_in[9];

  const int FIN  = 256;
  const int Nn   = in_sizes[0] / FIN;   // 10000
  const int E    = in_sizes[1] / 2;     // 320000
  const int Etot = E + Nn;              // + self-loops
  const int H1 = 8, C1 = 8, F1 = H1 * C1;   // 64
  const int H2 = 1, C2 = FIN;               // 256

  // workspace layout (floats), ~16.8 MB total
  float* ws    = (float*)d_ws;
  float* xl1   = ws;                          // Nn*F1
  float* asrc1 = xl1   + (size_t)Nn * F1;     // Nn*H1
  float* adst1 = asrc1 + (size_t)Nn * H1;
  float* maxv1 = adst1 + (size_t)Nn * H1;
  float* den1  = maxv1 + (size_t)Nn * H1;
  float* hbuf  = den1  + (size_t)Nn * H1;     // Nn*F1
  float* xl2   = hbuf  + (size_t)Nn * F1;     // Nn*C2
  float* asrc2 = xl2   + (size_t)Nn * C2;     // Nn
  float* adst2 = asrc2 + (size_t)Nn;
  float* maxv2 = adst2 + (size_t)Nn;
  float* den2  = maxv2 + (size_t)Nn;

  float* outp = (float*)d_out;
  const int TB = 256;
  auto blocks = [](long long work, int tb) { return (int)((work + tb - 1) / tb); };

  // ===== Layer 1 =====
  {
    int mG = ((Nn + 15) / 16 + GEMM_MT - 1) / GEMM_MT;
    int waves = mG * (F1 / 16);
    wmma_gemm_f32<<<blocks((long long)waves * 32, TB), TB, 0, stream>>>(
        x, W1, xl1, Nn, F1, FIN);
  }
  attn_coef<8, 8><<<blocks((long long)Nn * H1, TB), TB, 0, stream>>>(
      xl1, as1, ad1, asrc1, adst1, Nn);
  fill_f32<<<blocks((long long)Nn * H1, TB), TB, 0, stream>>>(maxv1, -INFINITY, Nn * H1);
  fill_f32<<<blocks((long long)Nn * H1, TB), TB, 0, stream>>>(den1, 0.f, Nn * H1);
  init_bias<64><<<blocks((long long)Nn * F1, TB), TB, 0, stream>>>(hbuf, b1, Nn);

  edge_max<8><<<blocks((long long)Etot * H1, TB), TB, 0, stream>>>(
      ei, E, Etot, asrc1, adst1, maxv1);
  edge_denom<8><<<blocks((long long)Etot * H1, TB), TB, 0, stream>>>(
      ei, E, Etot, asrc1, adst1, maxv1, den1);
  edge_aggregate<8, 8><<<blocks((long long)Etot * H1 * C1, TB), TB, 0, stream>>>(
      ei, E, Etot, asrc1, adst1, maxv1, den1, xl1, hbuf);

  elu_inplace<<<blocks((long long)Nn * F1, TB), TB, 0, stream>>>(hbuf, Nn * F1);

  // ===== Layer 2 =====
  {
    int mG = ((Nn + 15) / 16 + GEMM_MT - 1) / GEMM_MT;
    int waves = mG * (C2 / 16);
    wmma_gemm_f32<<<blocks((long long)waves * 32, TB), TB, 0, stream>>>(
        hbuf, W2, xl2, Nn, C2, F1);
  }
  attn_coef<1, 256><<<blocks((long long)Nn * H2, TB), TB, 0, stream>>>(
      xl2, as2, ad2, asrc2, adst2, Nn);
  fill_f32<<<blocks((long long)Nn * H2, TB), TB, 0, stream>>>(maxv2, -INFINITY, Nn * H2);
  fill_f32<<<blocks((long long)Nn * H2, TB), TB, 0, stream>>>(den2, 0.f, Nn * H2);
  init_bias<256><<<blocks((long long)Nn * C2, TB), TB, 0, stream>>>(outp, b2, Nn);

  edge_max<1><<<blocks((long long)Etot * H2, TB), TB, 0, stream>>>(
      ei, E, Etot, asrc2, adst2, maxv2);
  edge_denom<1><<<blocks((long long)Etot * H2, TB), TB, 0, stream>>>(
      ei, E, Etot, asrc2, adst2, maxv2, den2);
  edge_aggregate<1, 256><<<blocks((long long)Etot * H2 * C2, TB), TB, 0, stream>>>(
      ei, E, Etot, asrc2, adst2, maxv2, den2, xl2, outp);
}